// VariationalEncoderwithModel_1331439862312
// MI455X (gfx1250) — compile-verified
//
#include <hip/hip_runtime.h>

// GCN variational encoder for MI455X (gfx1250).
// - All h@W GEMMs use fp32 WMMA (V_WMMA_F32_16X16X4_F32): exact f32 precision,
//   ~4.9 GFLOP total. Each wave computes 4 row-tiles x one 16-col tile so each
//   B fragment feeds 4 WMMAs; K/OUT/RELU are template params -> fully unrolled,
//   immediate-offset loads, no per-element relu cndmask.
// - Edge aggregation is an L2-resident float-atomic scatter (working set
//   ~58 MB << 192 MB L2; HBM floor ~35 MB cold data at 23.3 TB/s).

typedef float v2f __attribute__((ext_vector_type(2)));
typedef float v8f __attribute__((ext_vector_type(8)));

static constexpr int HDIM = 128;
static constexpr int CDIM = 64;

// ---------------- degree / normalization ----------------
__global__ void deg_init_kernel(float* __restrict__ deg, int n) {
  int i = blockIdx.x * blockDim.x + threadIdx.x;
  if (i < n) deg[i] = 1.0f;  // "+1.0" self term in reference
}

__global__ void deg_accum_kernel(const int* __restrict__ dst, float* __restrict__ deg, int e) {
  int i = blockIdx.x * blockDim.x + threadIdx.x;
  if (i < e) atomicAdd(&deg[dst[i]], 1.0f);
}

__global__ void deg_finish_kernel(const float* __restrict__ deg, float* __restrict__ dinv,
                                  float* __restrict__ inv, int n) {
  int i = blockIdx.x * blockDim.x + threadIdx.x;
  if (i < n) {
    float d = deg[i];
    dinv[i] = rsqrtf(d);
    inv[i]  = 1.0f / d;
  }
}

// ---------------- embedding gather ----------------
__global__ void embed_kernel(const int* __restrict__ x, const float* __restrict__ embW,
                             float* __restrict__ h, int n) {
  int t = blockIdx.x * blockDim.x + threadIdx.x;
  if (t < n * HDIM) {
    int i = t >> 7;            // / HDIM
    int j = t & (HDIM - 1);
    h[t] = embW[x[i] * HDIM + j];
  }
}

// ---------------- fp32 WMMA GEMM: C[n x OUT] = (RELU?)A[n x K] @ W[K x OUT] ---
// Block = 128 threads = 4 wave32s. Wave w owns cols [blockIdx.y*64 + w*16, +16)
// for MT=4 row-tiles (64 rows) at blockIdx.x*64. K stepped by 4 per f32 WMMA.
// Fragments (ISA 7.12.2, 32-bit):
//   A 16x4 : lane<16: M=lane,   v0=K[k],  v1=K[k+1];  lane>=16: M=lane-16, K+=2
//   B 4x16 : lane<16: col=lane, rows k,k+1;           lane>=16: col=lane-16, rows k+2,k+3
//   C 16x16: vgpr v -> row v (lanes 0-15) / v+8 (lanes 16-31), col = lane&15
template <int K, int OUT, bool RELU, int MT>
__global__ void gemm_wmma_kernel(const float* __restrict__ A, const float* __restrict__ W,
                                 float* __restrict__ Cmat, int n) {
  const int lane = threadIdx.x & 31;
  const int wave = threadIdx.x >> 5;
  const int hi   = lane >> 4;        // half-wave: 0 or 1
  const int lm   = lane & 15;
  const int row0 = blockIdx.x * (16 * MT);
  const int col0 = blockIdx.y * 64 + wave * 16;

  // Fold the +2*hi K-offset into base pointers so unrolled offsets are immediates.
  const float* __restrict__ wbase = W + (size_t)(2 * hi) * OUT + col0 + lm;
  const float* arow[MT];
#pragma unroll
  for (int m = 0; m < MT; ++m) {
    int r = row0 + m * 16 + lm;
    if (r >= n) r = n - 1;           // clamp reads; stores are guarded below
    arow[m] = A + (size_t)r * K + 2 * hi;
  }

  v8f acc[MT];
#pragma unroll
  for (int m = 0; m < MT; ++m) acc[m] = (v8f){};

#pragma unroll
  for (int k = 0; k < K; k += 4) {
    v2f b;
    b.x = wbase[(size_t)k * OUT];
    b.y = wbase[(size_t)(k + 1) * OUT];
#pragma unroll
    for (int m = 0; m < MT; ++m) {
      v2f a = *reinterpret_cast<const v2f*>(arow[m] + k);  // 8B-aligned b64 load
      if (RELU) { a.x = fmaxf(a.x, 0.0f); a.y = fmaxf(a.y, 0.0f); }
      // 8 args: (neg_a, A, neg_b, B, c_mod, C, reuse_a, reuse_b)
      acc[m] = __builtin_amdgcn_wmma_f32_16x16x4_f32(false, a, false, b, (short)0,
                                                     acc[m], false, false);
    }
  }

  const int ccol = col0 + lm;
#pragma unroll
  for (int m = 0; m < MT; ++m) {
#pragma unroll
    for (int v = 0; v < 8; ++v) {
      int r = row0 + m * 16 + v + 8 * hi;
      if (r < n) Cmat[(size_t)r * OUT + ccol] = acc[m][v];
    }
  }
}

// ---------------- self term + bias: agg = hw * (1/deg)[row] + b[col] ----------
__global__ void agg_init_kernel(const float* __restrict__ hw, const float* __restrict__ inv,
                                const float* __restrict__ bias, float* __restrict__ agg,
                                int total, int shift, int OUT) {
  int t = blockIdx.x * blockDim.x + threadIdx.x;
  if (t < total) {
    int i = t >> shift;
    int j = t & (OUT - 1);
    agg[t] = hw[t] * inv[i] + bias[j];
  }
}

// ---------------- edge scatter: agg[dst] += hw[src] * dinv[src]*dinv[dst] -----
// One thread per (edge, 4-float column chunk): float4 gather + 4 f32 atomics.
__global__ void agg_edges_kernel(const int* __restrict__ src, const int* __restrict__ dst,
                                 const float* __restrict__ dinv, const float* __restrict__ hw,
                                 float* __restrict__ agg, int e, int OUT, int cshift) {
  unsigned t = blockIdx.x * blockDim.x + threadIdx.x;
  unsigned total = (unsigned)e << cshift;
  if (t >= total) return;
  int ei = (int)(t >> cshift);
  int c  = (int)((t & ((1u << cshift) - 1u)) << 2);
  int s = src[ei], d = dst[ei];
  float nrm = dinv[s] * dinv[d];
  const float4 hv = *reinterpret_cast<const float4*>(hw + (size_t)s * OUT + c);
  float* ap = agg + (size_t)d * OUT + c;
  atomicAdd(ap + 0, hv.x * nrm);
  atomicAdd(ap + 1, hv.y * nrm);
  atomicAdd(ap + 2, hv.z * nrm);
  atomicAdd(ap + 3, hv.w * nrm);
}

// ---------------- launch ----------------
extern "C" void kernel_launch(void* const* d_in, const int* in_sizes, int n_in,
                              void* d_out, int out_size, void* d_ws, size_t ws_size,
                              hipStream_t stream) {
  const int n = in_sizes[0];       // x is (N,1) int32
  const int e = in_sizes[1] / 2;   // edge_index is (2,E) int32

  const int*   xp   = (const int*)d_in[0];
  const int*   srcp = (const int*)d_in[1];
  const int*   dstp = srcp + e;
  const float* embW = (const float*)d_in[2];
  const float* W1   = (const float*)d_in[3];  const float* b1  = (const float*)d_in[4];
  const float* W2   = (const float*)d_in[5];  const float* b2  = (const float*)d_in[6];
  const float* W3   = (const float*)d_in[7];  const float* b3  = (const float*)d_in[8];
  const float* Wmu  = (const float*)d_in[9];  const float* bmu = (const float*)d_in[10];
  const float* Wls  = (const float*)d_in[11]; const float* bls = (const float*)d_in[12];

  // workspace layout: deg | dinv | inv | bufA (N*128) | bufB (N*128)
  size_t need = ((size_t)3 * n + 2 * (size_t)n * HDIM) * sizeof(float);
  if (ws_size < need) return;
  float* deg  = (float*)d_ws;
  float* dinv = deg + n;
  float* inv  = dinv + n;
  float* bufA = inv + n;
  float* bufB = bufA + (size_t)n * HDIM;

  float* mu_out = (float*)d_out;
  float* ls_out = mu_out + (size_t)n * CDIM;

  const int TB = 256;
  deg_init_kernel<<<(n + TB - 1) / TB, TB, 0, stream>>>(deg, n);
  deg_accum_kernel<<<(e + TB - 1) / TB, TB, 0, stream>>>(dstp, deg, e);
  deg_finish_kernel<<<(n + TB - 1) / TB, TB, 0, stream>>>(deg, dinv, inv, n);

  int embTot = n * HDIM;
  embed_kernel<<<(embTot + TB - 1) / TB, TB, 0, stream>>>(xp, embW, bufA, n);

  // GEMM launch helper (MT=4 row-tiles per wave => 64 rows per block)
  const int MT = 4;
  dim3 gH((n + 16 * MT - 1) / (16 * MT), HDIM / 64);  // OUT=128 -> y=2
  dim3 gC((n + 16 * MT - 1) / (16 * MT), CDIM / 64);  // OUT=64  -> y=1

  // aggregation launch helper
  auto aggregate = [&](int OUT, const float* hw, const float* b, float* aggout) {
    int total  = n * OUT;
    int shift  = (OUT == 128) ? 7 : 6;
    agg_init_kernel<<<(total + TB - 1) / TB, TB, 0, stream>>>(hw, inv, b, aggout,
                                                              total, shift, OUT);
    int cshift = (OUT == 128) ? 5 : 4;            // chunks = OUT/4
    long long etot = (long long)e << cshift;
    agg_edges_kernel<<<(unsigned)((etot + TB - 1) / TB), TB, 0, stream>>>(
        srcp, dstp, dinv, hw, aggout, e, OUT, cshift);
  };

  // conv1 (input: embedding, no relu); relu of each conv's output is applied
  // by the NEXT gemm's RELU template flag (gcn_conv only sees relu via input).
  gemm_wmma_kernel<HDIM, HDIM, false, MT><<<gH, 128, 0, stream>>>(bufA, W1, bufB, n);
  aggregate(HDIM, bufB, b1, bufA);

  gemm_wmma_kernel<HDIM, HDIM, true, MT><<<gH, 128, 0, stream>>>(bufA, W2, bufB, n);
  aggregate(HDIM, bufB, b2, bufA);

  gemm_wmma_kernel<HDIM, CDIM, true, MT><<<gC, 128, 0, stream>>>(bufA, W3, bufB, n);
  aggregate(CDIM, bufB, b3, bufA);

  gemm_wmma_kernel<CDIM, CDIM, true, MT><<<gC, 128, 0, stream>>>(bufA, Wmu, bufB, n);
  aggregate(CDIM, bufB, bmu, mu_out);

  gemm_wmma_kernel<CDIM, CDIM, true, MT><<<gC, 128, 0, stream>>>(bufA, Wls, bufB, n);
  aggregate(CDIM, bufB, bls, ls_out);
}